// SiSoMmsePicEqualizer_82626580840904
// MI455X (gfx1250) — compile-verified
//
#include <hip/hip_runtime.h>
#include <hip/hip_bf16.h>

// CDNA5 / gfx1250: wave32, WMMA f32 16x16x4 for the Gram matrix z^H z,
// async global->LDS copy (GLOBAL_LOAD_ASYNC_TO_LDS_B128) for the S tile.

typedef __attribute__((ext_vector_type(2))) float v2f;
typedef __attribute__((ext_vector_type(8))) float v8f;
typedef __attribute__((ext_vector_type(4))) int   vi4;

#define WSYNC() __builtin_amdgcn_wave_barrier()

#define EPS_F 1e-4f
#define INV_SQRT10 0.31622776601683794f

#if defined(__has_builtin)
#if __has_builtin(__builtin_amdgcn_global_load_async_to_lds_b128) && \
    __has_builtin(__builtin_amdgcn_s_wait_asynccnt)
#define HAVE_ASYNC_LDS 1
#endif
#endif
#ifndef HAVE_ASYNC_LDS
#define HAVE_ASYNC_LDS 0
#endif

// QAM16 Gray constellation (Sionna convention), unit average energy.
__device__ __constant__ float CRE[16] = {
     1*INV_SQRT10,  1*INV_SQRT10,  3*INV_SQRT10,  3*INV_SQRT10,
     1*INV_SQRT10,  1*INV_SQRT10,  3*INV_SQRT10,  3*INV_SQRT10,
    -1*INV_SQRT10, -1*INV_SQRT10, -3*INV_SQRT10, -3*INV_SQRT10,
    -1*INV_SQRT10, -1*INV_SQRT10, -3*INV_SQRT10, -3*INV_SQRT10 };
__device__ __constant__ float CIM[16] = {
     1*INV_SQRT10,  3*INV_SQRT10,  1*INV_SQRT10,  3*INV_SQRT10,
    -1*INV_SQRT10, -3*INV_SQRT10, -1*INV_SQRT10, -3*INV_SQRT10,
     1*INV_SQRT10,  3*INV_SQRT10,  1*INV_SQRT10,  3*INV_SQRT10,
    -1*INV_SQRT10, -3*INV_SQRT10, -1*INV_SQRT10, -3*INV_SQRT10 };

struct __align__(16) WaveScratch {
    float sm[512];          // 16x16 complex (re,im interleaved); reused as 16x16 f32 D
    float zsr[16][9];       // Re(L^{-1}[h|y])
    float zsi[16][9];       // Im
    float gmr[64], gmi[64]; // 8x8 complex Gram
    float augr[128], augi[128]; // 8x16 complex augmented [A | I]
    float ymfr[8], ymfi[8];
    float xhr[8], xhi[8], vx[8];
    float gxr[8], gxi[8];
};

#define SRE(i,j) W.sm[((((i)<<4)+(j))<<1)]
#define SIM(i,j) W.sm[((((i)<<4)+(j))<<1)+1]

__global__ __launch_bounds__(256)
void mmse_pic_kernel(const float* __restrict__ yg,
                     const float* __restrict__ hg,
                     const float* __restrict__ priorg,
                     const float* __restrict__ sg,
                     float* __restrict__ out, int Btot)
{
    __shared__ WaveScratch wsm[8];
    const int lane = threadIdx.x & 31;
    const int wave = threadIdx.x >> 5;
    const int b = blockIdx.x * 8 + wave;
    if (b >= Btot) return;               // wave-uniform exit: EXEC stays full
    WaveScratch& W = wsm[wave];

    // ---------------- load S[b] (16x16 complex64) into LDS ----------------
#if HAVE_ASYNC_LDS
    {
        const vi4* src4 = (const vi4*)(sg + (size_t)b * 512);
        vi4*       dst4 = (vi4*)W.sm;
        #pragma unroll
        for (int q = 0; q < 4; ++q) {
            __builtin_amdgcn_global_load_async_to_lds_b128(
                (__attribute__((address_space(1))) vi4*)(src4 + (q * 32 + lane)),
                (__attribute__((address_space(3))) vi4*)(dst4 + (q * 32 + lane)),
                0, 0);
        }
    }
#else
    {
        const float4* s4 = (const float4*)(sg + (size_t)b * 512);
        float4* d4 = (float4*)W.sm;
        #pragma unroll
        for (int q = 0; q < 4; ++q) d4[q * 32 + lane] = s4[q * 32 + lane];
    }
#endif

    // ---------------- prior -> (x_hat, var_x) moments, lanes 0..7 ----------
    // (independent of S: overlaps with the async S copy)
    if (lane < 8) {
        const int k = lane;
        const float* pp = priorg + ((size_t)b * 8 + k) * 4;
        const float p0 = pp[0], p1 = pp[1], p2 = pp[2], p3 = pp[3];
        float lg[16], mx = -1e30f;
        #pragma unroll
        for (int c = 0; c < 16; ++c) {
            float a0 = (c & 8) ? p0 : -p0;   // bit n=0 is the MSB
            float a1 = (c & 4) ? p1 : -p1;
            float a2 = (c & 2) ? p2 : -p2;
            float a3 = (c & 1) ? p3 : -p3;
            lg[c] = 0.5f * (a0 + a1 + a2 + a3);
            mx = fmaxf(mx, lg[c]);
        }
        float pe[16], sum = 0.f, mr = 0.f, mi = 0.f;
        #pragma unroll
        for (int c = 0; c < 16; ++c) {
            pe[c] = __expf(lg[c] - mx);
            sum += pe[c]; mr += pe[c] * CRE[c]; mi += pe[c] * CIM[c];
        }
        const float inv = 1.0f / sum;
        mr *= inv; mi *= inv;
        float var = 0.f;
        #pragma unroll
        for (int c = 0; c < 16; ++c) {
            float dr = CRE[c] - mr, di = CIM[c] - mi;
            var += pe[c] * (dr * dr + di * di);
        }
        var *= inv;
        W.xhr[k] = mr; W.xhi[k] = mi; W.vx[k] = var;
    }
#if HAVE_ASYNC_LDS
    __builtin_amdgcn_s_wait_asynccnt(0);   // S tile now resident in LDS
#endif
    WSYNC();

    // ---------------- Cholesky: S = L L^H  (in place; diag stores 1/L[k][k])
    for (int k = 0; k < 16; ++k) {
        const float d = sqrtf(fmaxf(SRE(k, k), 1e-20f));
        const float invd = 1.0f / d;
        WSYNC();
        if (lane == k) { SRE(k, k) = invd; SIM(k, k) = 0.0f; }
        if (lane > k && lane < 16) { SRE(lane, k) *= invd; SIM(lane, k) *= invd; }
        WSYNC();
        const int i = lane & 15;
        for (int j = k + 1 + (lane >> 4); j < 16; j += 2) {
            if (i >= j) {
                const float ar = SRE(i, k), ai = SIM(i, k);
                const float br = SRE(j, k), bi = SIM(j, k);
                SRE(i, j) -= ar * br + ai * bi;   // s[i][j] -= L[i][k]*conj(L[j][k])
                SIM(i, j) -= ai * br - ar * bi;
            }
        }
        WSYNC();
    }

    // ---------------- forward solve z = L^{-1} [h | y], lanes 0..8 ---------
    if (lane < 9) {
        const int c = lane;
        float br_[16], bi_[16];
        if (c < 8) {
            const float2* h2 = (const float2*)hg;
            #pragma unroll
            for (int i = 0; i < 16; ++i) {
                float2 v = h2[((size_t)b * 16 + i) * 8 + c];
                br_[i] = v.x; bi_[i] = v.y;
            }
        } else {
            const float2* y2 = (const float2*)yg;
            #pragma unroll
            for (int i = 0; i < 16; ++i) {
                float2 v = y2[(size_t)b * 16 + i];
                br_[i] = v.x; bi_[i] = v.y;
            }
        }
        float zr[16], zi[16];
        #pragma unroll
        for (int i = 0; i < 16; ++i) {
            float ar = br_[i], ai = bi_[i];
            #pragma unroll
            for (int j = 0; j < i; ++j) {
                const float lr = SRE(i, j), li = SIM(i, j); // uniform -> LDS broadcast
                ar -= lr * zr[j] - li * zi[j];
                ai -= lr * zi[j] + li * zr[j];
            }
            const float invd = SRE(i, i);
            zr[i] = ar * invd; zi[i] = ai * invd;
            W.zsr[i][c] = zr[i]; W.zsi[i][c] = zi[i];
        }
    }
    WSYNC();

    // ---------------- y_mf = z^H w, lanes 0..7 -----------------------------
    if (lane < 8) {
        const int k = lane;
        float sr = 0.f, si = 0.f;
        #pragma unroll
        for (int m = 0; m < 16; ++m) {
            const float zr = W.zsr[m][k], zi = W.zsi[m][k];
            const float wr = W.zsr[m][8], wi = W.zsi[m][8];
            sr += zr * wr + zi * wi;     // conj(z)*w
            si += zr * wi - zi * wr;
        }
        W.ymfr[k] = sr; W.ymfi[k] = si;
    }
    WSYNC();

    // ---------------- Gram via WMMA: D = P^T P, P = [Re z | Im z] ----------
    // 16x16x4 f32 layouts: A lane L: {A[L&15, 2*hi], A[L&15, 2*hi+1]};
    // B lane L: {B[2*hi, L&15], B[2*hi+1, L&15]}  => identical registers here.
    {
        const int col = lane & 15;
        const int hi  = lane >> 4;
        v8f acc = {0.f, 0.f, 0.f, 0.f, 0.f, 0.f, 0.f, 0.f};
        #pragma unroll
        for (int q = 0; q < 4; ++q) {
            const int r0 = 4 * q + 2 * hi;
            v2f ab;
            ab.x = (col < 8) ? W.zsr[r0][col]     : W.zsi[r0][col - 8];
            ab.y = (col < 8) ? W.zsr[r0 + 1][col] : W.zsi[r0 + 1][col - 8];
            acc = __builtin_amdgcn_wmma_f32_16x16x4_f32(
                false, ab, false, ab, (short)0, acc, false, false);
        }
        float* dmat = W.sm;              // L no longer needed: reuse as 16x16 f32
        WSYNC();
        #pragma unroll
        for (int v = 0; v < 8; ++v) dmat[(v + 8 * hi) * 16 + col] = acc[v];
    }
    WSYNC();
    // combine blocks: G_re = D00 + D11, G_im = D01 - D10
    {
        const int k  = lane >> 2;
        const int l0 = (lane & 3) * 2;
        const float* dmat = W.sm;
        #pragma unroll
        for (int t = 0; t < 2; ++t) {
            const int l = l0 + t;
            W.gmr[k * 8 + l] = dmat[k * 16 + l] + dmat[(k + 8) * 16 + (l + 8)];
            W.gmi[k * 8 + l] = dmat[k * 16 + (l + 8)] - dmat[(k + 8) * 16 + l];
        }
    }
    WSYNC();

    // ---------------- 4 MMSE-PIC iterations --------------------------------
    const int ac = lane & 15;
    const int ah = lane >> 4;
    for (int it = 0; it < 4; ++it) {
        // build augmented [A | I], A = G*diag(var) + I
        #pragma unroll
        for (int rr = 0; rr < 4; ++rr) {
            const int r = 2 * rr + ah;
            float ar, ai;
            if (ac < 8) {
                const float v = W.vx[ac];
                ar = W.gmr[r * 8 + ac] * v;
                ai = W.gmi[r * 8 + ac] * v;
                if (r == ac) ar += 1.0f;
            } else {
                ar = (ac - 8 == r) ? 1.0f : 0.0f; ai = 0.0f;
            }
            W.augr[r * 16 + ac] = ar; W.augi[r * 16 + ac] = ai;
        }
        WSYNC();
        // Gauss-Jordan (A = I + G*diag(var): diagonally dominant-ish, no pivoting)
        for (int p = 0; p < 8; ++p) {
            const float pr = W.augr[p * 16 + p], pi = W.augi[p * 16 + p];
            const float iden = 1.0f / (pr * pr + pi * pi);
            const float ipr = pr * iden, ipi = -pi * iden;
            WSYNC();
            if (ah == 0) {                    // scale pivot row (16 cols / 16 lanes)
                const float xr = W.augr[p * 16 + ac], xi = W.augi[p * 16 + ac];
                W.augr[p * 16 + ac] = xr * ipr - xi * ipi;
                W.augi[p * 16 + ac] = xr * ipi + xi * ipr;
            }
            WSYNC();
            float fr[4], fi[4];
            #pragma unroll
            for (int rr = 0; rr < 4; ++rr) {
                const int r = 2 * rr + ah;
                fr[rr] = W.augr[r * 16 + p]; fi[rr] = W.augi[r * 16 + p];
            }
            const float prr = W.augr[p * 16 + ac], pri = W.augi[p * 16 + ac];
            WSYNC();
            #pragma unroll
            for (int rr = 0; rr < 4; ++rr) {
                const int r = 2 * rr + ah;
                if (r != p) {
                    W.augr[r * 16 + ac] -= fr[rr] * prr - fi[rr] * pri;
                    W.augi[r * 16 + ac] -= fr[rr] * pri + fi[rr] * prr;
                }
            }
            WSYNC();
        }
        // gx = G @ x_hat
        if (lane < 8) {
            const int k = lane;
            float sr = 0.f, si = 0.f;
            #pragma unroll
            for (int j = 0; j < 8; ++j) {
                const float gr = W.gmr[k * 8 + j], gi = W.gmi[k * 8 + j];
                const float xr = W.xhr[j],        xi = W.xhi[j];
                sr += gr * xr - gi * xi;
                si += gr * xi + gi * xr;
            }
            W.gxr[k] = sr; W.gxi[k] = si;
        }
        WSYNC();
        // x_hat, var update
        if (lane < 8) {
            const int k = lane;
            float dcr = 0.f, dci = 0.f, rr_ = 0.f, ri_ = 0.f;
            #pragma unroll
            for (int l = 0; l < 8; ++l) {
                const float Ar = W.augr[k * 16 + 8 + l], Ai = W.augi[k * 16 + 8 + l];
                const float gr = W.gmr[l * 8 + k],       gi = W.gmi[l * 8 + k];
                dcr += Ar * gr - Ai * gi;                 // (A^{-1} G)_{kk}
                dci += Ar * gi + Ai * gr;
                const float tr = W.ymfr[l] - W.gxr[l];
                const float ti = W.ymfi[l] - W.gxi[l];
                rr_ += Ar * tr - Ai * ti;                 // A^{-1}(y_mf - Gx)
                ri_ += Ar * ti + Ai * tr;
            }
            const float mu  = dcr;
            const float imu = 1.0f / mu;
            const float xr = W.xhr[k], xi = W.xhi[k];
            const float nr = (rr_ + xr * dcr - xi * dci) * imu;
            const float ni = (ri_ + xr * dci + xi * dcr) * imu;
            const float vold = W.vx[k];
            W.xhr[k] = nr; W.xhi[k] = ni;
            W.vx[k]  = mu / fmaxf(1.0f - vold * mu, EPS_F);
        }
        WSYNC();
    }

    // ---------------- outputs: x_hat (complex64) then no_eff (f32) ---------
    if (lane < 8) {
        const int k = lane;
        out[((size_t)b * 8 + k) * 2]     = W.xhr[k];
        out[((size_t)b * 8 + k) * 2 + 1] = W.xhi[k];
        out[(size_t)Btot * 16 + (size_t)b * 8 + k] = 1.0f / W.vx[k];
    }
}

extern "C" void kernel_launch(void* const* d_in, const int* in_sizes, int n_in,
                              void* d_out, int out_size, void* d_ws, size_t ws_size,
                              hipStream_t stream) {
    (void)n_in; (void)out_size; (void)d_ws; (void)ws_size;
    const float* y     = (const float*)d_in[0];   // [B,16]    complex64
    const float* h     = (const float*)d_in[1];   // [B,16,8]  complex64
    const float* prior = (const float*)d_in[2];   // [B,8,4]   float32
    const float* s     = (const float*)d_in[3];   // [B,16,16] complex64
    float* out = (float*)d_out;                   // [B,8] cplx + [B,8] f32

    const int Btot = in_sizes[2] / 32;            // prior is B*8*4 floats
    const int blocks = (Btot + 7) / 8;            // 8 waves/block, 1 item/wave
    mmse_pic_kernel<<<blocks, 256, 0, stream>>>(y, h, prior, s, out, Btot);
}